// PCLinearClassifierV2_32744830665380
// MI455X (gfx1250) — compile-verified
//
#include <hip/hip_runtime.h>

// ---------------------------------------------------------------------------
// PC network forward == pure feedforward (errors are identically zero at init,
// so the 5 inference iterations are exact no-ops). Three chained GEMMs:
//   C = relu(A) @ W^T + b   with A:(M,K) row-major, W:(N,K) row-major.
// fp32 WMMA (V_WMMA_F32_16X16X4_F32) for exact dtype match with the fp32
// reference at matrix-pipe rate. Staging uses CDNA5 async global->LDS DMA
// (GLOBAL_LOAD_ASYNC_TO_LDS_B128, ASYNCcnt) with double-buffered LDS; the
// inner loop software-pipelines LDS fragment loads against WMMAs. ReLU is a
// single raw v_max_num_f32 per element (inline asm defeats the compiler's
// canonicalize insertion); bias is folded into the accumulator init so the
// epilogue is store-only.
// ---------------------------------------------------------------------------

typedef float v2f __attribute__((ext_vector_type(2)));
typedef float v8f __attribute__((ext_vector_type(8)));

#define BM 128
#define BN 128
#define BK 32
#define LDT 36   // padded LDS row stride (floats): 16B-aligned rows, conflict-free b64 reads

__device__ __forceinline__ void async_copy_b128(unsigned lds_off, const void* gptr) {
    // VDST = LDS byte address, VADDR = 64-bit global address, saddr = off
    asm volatile("global_load_async_to_lds_b128 %0, %1, off"
                 :: "v"(lds_off), "v"(gptr)
                 : "memory");
}

__device__ __forceinline__ void wait_async0() {
    asm volatile("s_wait_asynccnt 0x0" ::: "memory");
}

__device__ __forceinline__ unsigned lds_off_of(const void* p) {
    // low 32 bits of the flat LDS-aperture address == allocation-relative offset
    return (unsigned)(unsigned long long)(p);
}

__device__ __forceinline__ float relu1(float x) {
    // raw IEEE maximumNumber(x, 0): exactly max(x,0) for non-NaN inputs,
    // one VALU op, no canonicalize
    float r;
    asm("v_max_num_f32 %0, %1, 0" : "=v"(r) : "v"(x));
    return r;
}

__global__ __launch_bounds__(256)
void pc_gemm_relu_wt(const float* __restrict__ A,    // (M,K); relu applied on fragment load
                     const float* __restrict__ W,    // (N,K)
                     const float* __restrict__ bias, // (N)
                     float* __restrict__ C,          // (M,N), stored raw (no relu)
                     int M, int N, int K)
{
    __shared__ __align__(16) float As[2][BM * LDT];
    __shared__ __align__(16) float Ws[2][BM * LDT];

    const int tid    = threadIdx.x;
    const int lane   = tid & 31;
    const int wid    = tid >> 5;
    const int mw     = wid >> 2;        // 0..1  : 64-row half of the C tile
    const int nw     = wid & 3;         // 0..3  : 32-col slice of the C tile
    const int laneM  = lane & 15;       // row/col within a 16x16 fragment
    const int laneHi = lane >> 4;       // 0/1   : K-pair select (A/B), +8 rows (C/D)

    const int mBlock = blockIdx.x * BM;
    const int nBlock = blockIdx.y * BN;

    // Bias folded into the accumulators: every component of acc[i][j] maps to
    // the same output column (col = colBase + lane%16), so broadcast-init.
    v8f acc[4][2];
#pragma unroll
    for (int j = 0; j < 2; ++j) {
        const int col = nBlock + nw * 32 + j * 16 + laneM;
        const float bvj = (col < N) ? bias[col] : 0.0f;
#pragma unroll
        for (int i = 0; i < 4; ++i)
            acc[i][j] = (v8f){bvj, bvj, bvj, bvj, bvj, bvj, bvj, bvj};
    }

    // Async staging map: 256 threads x 4 iters x b128 moves one 128x32 fp32 tile.
    const int rowStage = tid >> 3;        // 0..31
    const int colStage = (tid & 7) * 4;   // 0,4,...,28

    const int nChunks = K / BK;

    // ---- issue async DMA for chunk 0 into buffer 0 ----
#pragma unroll
    for (int it = 0; it < 4; ++it) {
        const int r = it * 32 + rowStage;
        async_copy_b128(lds_off_of(&As[0][r * LDT + colStage]),
                        A + (size_t)(mBlock + r) * K + colStage);
        int wr = nBlock + r;
        if (wr > N - 1) wr = N - 1;     // ragged N: duplicated rows feed never-stored columns
        async_copy_b128(lds_off_of(&Ws[0][r * LDT + colStage]),
                        W + (size_t)wr * K + colStage);
    }
    wait_async0();
    __syncthreads();

    for (int c = 0; c < nChunks; ++c) {
        const int buf = c & 1;

        // ---- issue async DMA for chunk c+1 into the other buffer ----
        if (c + 1 < nChunks) {
            const int k0 = (c + 1) * BK;
#pragma unroll
            for (int it = 0; it < 4; ++it) {
                const int r = it * 32 + rowStage;
                async_copy_b128(lds_off_of(&As[buf ^ 1][r * LDT + colStage]),
                                A + (size_t)(mBlock + r) * K + k0 + colStage);
                int wr = nBlock + r;
                if (wr > N - 1) wr = N - 1;
                async_copy_b128(lds_off_of(&Ws[buf ^ 1][r * LDT + colStage]),
                                W + (size_t)wr * K + k0 + colStage);
            }
        }

        // ---- compute chunk c: software-pipelined frags + 64 WMMAs ----
        const float* __restrict__ Ab = &As[buf][0];
        const float* __restrict__ Wb = &Ws[buf][0];

        v2f a[2][4], b[2][2];
        {
            const int kf = 2 * laneHi;                 // kk = 0
#pragma unroll
            for (int i = 0; i < 4; ++i)
                a[0][i] = *(const v2f*)(Ab + (mw * 64 + i * 16 + laneM) * LDT + kf);
#pragma unroll
            for (int j = 0; j < 2; ++j)
                b[0][j] = *(const v2f*)(Wb + (nw * 32 + j * 16 + laneM) * LDT + kf);
        }

#pragma unroll
        for (int kk = 0; kk < 8; ++kk) {
            const int cur = kk & 1;
            if (kk + 1 < 8) {                          // prefetch next K-step's frags
                const int kf = (kk + 1) * 4 + 2 * laneHi;
#pragma unroll
                for (int i = 0; i < 4; ++i)
                    a[cur ^ 1][i] = *(const v2f*)(Ab + (mw * 64 + i * 16 + laneM) * LDT + kf);
#pragma unroll
                for (int j = 0; j < 2; ++j)
                    b[cur ^ 1][j] = *(const v2f*)(Wb + (nw * 32 + j * 16 + laneM) * LDT + kf);
            }

            v2f ar[4];                                 // relu(A): 1 v_max_num_f32 each
#pragma unroll
            for (int i = 0; i < 4; ++i) {
                ar[i][0] = relu1(a[cur][i][0]);
                ar[i][1] = relu1(a[cur][i][1]);
            }

#pragma unroll
            for (int i = 0; i < 4; ++i)
#pragma unroll
                for (int j = 0; j < 2; ++j)
                    acc[i][j] = __builtin_amdgcn_wmma_f32_16x16x4_f32(
                        false, ar[i], false, b[cur][j], (short)0, acc[i][j],
                        false, false);
        }

        wait_async0();      // next chunk's DMA has landed in the other buffer
        __syncthreads();    // all waves done reading this buffer
    }

    // Epilogue: pure stores. C/D layout -> row = r + 8*laneHi, col = laneM.
#pragma unroll
    for (int i = 0; i < 4; ++i) {
#pragma unroll
        for (int j = 0; j < 2; ++j) {
            const int col = nBlock + nw * 32 + j * 16 + laneM;
            if (col < N) {
#pragma unroll
                for (int r = 0; r < 8; ++r) {
                    const int row = mBlock + mw * 64 + i * 16 + r + 8 * laneHi;
                    C[(size_t)row * N + col] = acc[i][j][r];
                }
            }
        }
    }
}

extern "C" void kernel_launch(void* const* d_in, const int* in_sizes, int n_in,
                              void* d_out, int out_size, void* d_ws, size_t ws_size,
                              hipStream_t stream) {
    // setup_inputs order: obs, W0, b0, W1, b1, W2, b2
    const float* obs = (const float*)d_in[0];
    const float* W0  = (const float*)d_in[1];
    const float* b0  = (const float*)d_in[2];
    const float* W1  = (const float*)d_in[3];
    const float* b1  = (const float*)d_in[4];
    const float* W2  = (const float*)d_in[5];
    const float* b2  = (const float*)d_in[6];
    float* out = (float*)d_out;

    float* x2 = (float*)d_ws;             // 1024 x 1024
    float* x1 = x2 + (size_t)1024 * 1024; // 1024 x 2048  (12 MB total scratch)

    dim3 blk(256);
    // x2 = relu(obs) @ W2^T + b2   (M=1024, N=1024, K=4096)
    pc_gemm_relu_wt<<<dim3(8, 8),  blk, 0, stream>>>(obs, W2, b2, x2, 1024, 1024, 4096);
    // x1 = relu(x2) @ W1^T + b1    (M=1024, N=2048, K=1024)
    pc_gemm_relu_wt<<<dim3(8, 16), blk, 0, stream>>>(x2, W1, b1, x1, 1024, 2048, 1024);
    // x0 = relu(x1) @ W0^T + b0    (M=1024, N=1000, K=2048)
    pc_gemm_relu_wt<<<dim3(8, 8),  blk, 0, stream>>>(x1, W0, b0, out, 1024, 1000, 2048);
}